// DataTaskBipartiteLayer_7095285973618
// MI455X (gfx1250) — compile-verified
//
#include <hip/hip_runtime.h>
#include <math.h>

// ---------------------------------------------------------------------------
// Problem constants (from reference)
// ---------------------------------------------------------------------------
#define NT 200000
#define ND 100000
#define EDGES 1000000
#define TF 12
#define DF 5
#define EF 3
#define HC 64

typedef __attribute__((ext_vector_type(2))) float v2f;
typedef __attribute__((ext_vector_type(8))) float v8f;

// ---------------------------------------------------------------------------
// Order-preserving float <-> unsigned encoding for atomicMax on floats
// ---------------------------------------------------------------------------
__device__ __forceinline__ unsigned fenc(float f) {
  unsigned u = __float_as_uint(f);
  return (u & 0x80000000u) ? ~u : (u | 0x80000000u);
}
__device__ __forceinline__ float fdec(unsigned u) {
  unsigned v = (u & 0x80000000u) ? (u & 0x7FFFFFFFu) : ~u;
  return __uint_as_float(v);
}

// ---------------------------------------------------------------------------
// wvec[k] = sum_c M[c*K + k] * a[c]   (M is [64,K] row-major, a is [64])
// Used for Wd^T @ a_d (length ind) and We^T @ a_e (length 3).
// ---------------------------------------------------------------------------
__global__ void matTvec_kernel(const float* __restrict__ M,
                               const float* __restrict__ a,
                               float* __restrict__ out, int K) {
  int k = threadIdx.x;
  if (k >= K) return;
  float s = 0.f;
  for (int c = 0; c < HC; ++c) s += M[c * K + k] * a[c];
  out[k] = s;
}

// ---------------------------------------------------------------------------
// Small-K GEMM: Y[n][c] = sum_k X[n][k]*W[c][k] (+bias). One thread per (n,c).
// K is 5 or 12: memory-bound on the Y write; W lives in L2/L0.
// ---------------------------------------------------------------------------
__global__ __launch_bounds__(256) void gemm_smallk_kernel(
    const float* __restrict__ X, const float* __restrict__ W,
    const float* __restrict__ bias, float* __restrict__ Y,
    int N, int K) {
  size_t tid = (size_t)blockIdx.x * blockDim.x + threadIdx.x;
  if (tid >= (size_t)N * HC) return;
  int c = (int)(tid & 63);
  size_t n = tid >> 6;
  const float* x = X + n * K;
  const float* w = W + (size_t)c * K;
  float s = bias ? bias[c] : 0.f;
  for (int k = 0; k < K; ++k) s += x[k] * w[k];
  Y[tid] = s;
}

// ---------------------------------------------------------------------------
// K=64 GEMM via V_WMMA_F32_16X16X4_F32 (full fp32 precision).
// Y[n][c] = sum_{k<64} X[n][k] * W[c][k] (+bias)   (i.e. Y = X @ W^T)
// 8 waves per block; wave w -> row tile (2*blockIdx + w/4), col tile (w%4).
// N must be a multiple of 32 (NT=200000, ND=100000 both are).
// A 16x4 layout: lane<16 holds K={0,1}, lane>=16 holds K={2,3} (ISA 7.12.2).
// B 4x16: lanes 0-15 hold K-half 0, lanes 16-31 hold K-half 1.
// D 16x16: VGPR v: lane<16 -> row v, lane>=16 -> row v+8.
// ---------------------------------------------------------------------------
__global__ __launch_bounds__(256) void gemm64_wmma_kernel(
    const float* __restrict__ X, const float* __restrict__ W,
    const float* __restrict__ bias, float* __restrict__ Y, int N) {
  int wave = threadIdx.x >> 5;
  int lane = threadIdx.x & 31;
  int row0 = (blockIdx.x * 2 + (wave >> 2)) * 16;
  int col0 = (wave & 3) * 16;
  if (row0 >= N) return;                    // wave-uniform: EXEC stays all-ones
  int half = lane >> 4;                     // K sub-pair select
  int l    = lane & 15;
  const float* xrow = X + (size_t)(row0 + l) * HC + half * 2;
  const float* wrow = W + (size_t)(col0 + l) * HC + half * 2;
  v8f acc = {};
#pragma unroll
  for (int s = 0; s < 16; ++s) {            // 16 steps of K=4 -> K=64
    v2f a, b;
    a.x = xrow[s * 4 + 0];
    a.y = xrow[s * 4 + 1];
    b.x = wrow[s * 4 + 0];
    b.y = wrow[s * 4 + 1];
    acc = __builtin_amdgcn_wmma_f32_16x16x4_f32(
        false, a, false, b, (short)0, acc, false, false);
  }
  float bv = bias ? bias[col0 + l] : 0.f;
#pragma unroll
  for (int v = 0; v < 8; ++v) {
    int r = row0 + v + half * 8;
    Y[(size_t)r * HC + col0 + l] = acc[v] + bv;
  }
}

// ---------------------------------------------------------------------------
// out[n] = dot(X[n, :K], v)
// ---------------------------------------------------------------------------
__global__ __launch_bounds__(256) void dot_rows_kernel(
    const float* __restrict__ X, const float* __restrict__ v,
    float* __restrict__ out, int N, int K) {
  int n = blockIdx.x * blockDim.x + threadIdx.x;
  if (n >= N) return;
  const float* x = X + (size_t)n * K;
  float s = 0.f;
  for (int k = 0; k < K; ++k) s += x[k] * v[k];
  out[n] = s;
}

__global__ __launch_bounds__(256) void init_softmax_kernel(
    unsigned* __restrict__ amax, float* __restrict__ den, int N) {
  int n = blockIdx.x * blockDim.x + threadIdx.x;
  if (n >= N) return;
  amax[n] = fenc(-INFINITY);
  den[n] = 0.f;
}

// Pass 1: per-edge logit + leaky_relu(0.2), stash it, segment-max via atomicMax
__global__ __launch_bounds__(256) void edge_attn_max_kernel(
    const int* __restrict__ src, const int* __restrict__ dst,
    const float* __restrict__ ea, const float* __restrict__ al_s,
    const float* __restrict__ al_d, const float* __restrict__ wea,
    float* __restrict__ albuf, unsigned* __restrict__ amax, int E) {
  int e = blockIdx.x * blockDim.x + threadIdx.x;
  if (e >= E) return;
  const float* a = ea + (size_t)e * EF;
  float al = al_s[src[e]] + al_d[dst[e]] +
             a[0] * wea[0] + a[1] * wea[1] + a[2] * wea[2];
  al = (al > 0.f) ? al : 0.2f * al;
  albuf[e] = al;
  atomicMax(&amax[dst[e]], fenc(al));
}

// amax decode (+ replace -inf of empty segments with 0, like the reference)
__global__ __launch_bounds__(256) void amax_decode_kernel(unsigned* amax, int N) {
  int n = blockIdx.x * blockDim.x + threadIdx.x;
  if (n >= N) return;
  float f = fdec(amax[n]);
  if (!__builtin_isfinite(f)) f = 0.f;
  ((float*)amax)[n] = f;
}

// Pass 2: ex = exp(al - amax[dst]); albuf <- ex; den[dst] += ex
__global__ __launch_bounds__(256) void edge_exp_kernel(
    const int* __restrict__ dst, float* __restrict__ albuf,
    const float* __restrict__ amaxf, float* __restrict__ den, int E) {
  int e = blockIdx.x * blockDim.x + threadIdx.x;
  if (e >= E) return;
  float ex = expf(albuf[e] - amaxf[dst[e]]);
  albuf[e] = ex;
  unsafeAtomicAdd(&den[dst[e]], ex);        // global_atomic_add_f32
}

// Pass 3: out[dst] += xs[src] * (ex / (den[dst] + 1e-16)); 16 threads/edge x f4
__global__ __launch_bounds__(256) void edge_agg_kernel(
    const int* __restrict__ src, const int* __restrict__ dst,
    const float* __restrict__ ex, const float* __restrict__ den,
    const float* __restrict__ xs, float* __restrict__ out, int E) {
  size_t tid = (size_t)blockIdx.x * blockDim.x + threadIdx.x;
  if (tid >= (size_t)E * 16) return;
  int e = (int)(tid >> 4);
  int q = (int)(tid & 15) * 4;
  int s = src[e], d = dst[e];
  float alpha = ex[e] / (den[d] + 1e-16f);
  const float4 x = *(const float4*)(xs + (size_t)s * HC + q);
  float* o = out + (size_t)d * HC + q;
  unsafeAtomicAdd(o + 0, x.x * alpha);
  unsafeAtomicAdd(o + 1, x.y * alpha);
  unsafeAtomicAdd(o + 2, x.z * alpha);
  unsafeAtomicAdd(o + 3, x.w * alpha);
}

// leaky_relu(0.01) + LayerNorm(64) in place; one wave32 per row (2 ch/lane)
__global__ __launch_bounds__(256) void act_ln_kernel(
    float* __restrict__ x, const float* __restrict__ g,
    const float* __restrict__ b, int N) {
  int lane = threadIdx.x & 31;
  int row = (int)(((size_t)blockIdx.x * blockDim.x + threadIdx.x) >> 5);
  if (row >= N) return;
  float* r = x + (size_t)row * HC;
  float v0 = r[lane], v1 = r[lane + 32];
  v0 = (v0 > 0.f) ? v0 : 0.01f * v0;
  v1 = (v1 > 0.f) ? v1 : 0.01f * v1;
  float s = v0 + v1, s2 = v0 * v0 + v1 * v1;
#pragma unroll
  for (int off = 16; off; off >>= 1) {
    s  += __shfl_xor(s, off, 32);
    s2 += __shfl_xor(s2, off, 32);
  }
  float mu  = s * (1.f / 64.f);
  float var = s2 * (1.f / 64.f) - mu * mu;
  float inv = rsqrtf(var + 1e-5f);
  r[lane]      = (v0 - mu) * inv * g[lane]      + b[lane];
  r[lane + 32] = (v1 - mu) * inv * g[lane + 32] + b[lane + 32];
}

// ---------------------------------------------------------------------------
// Host orchestration
// ---------------------------------------------------------------------------
extern "C" void kernel_launch(void* const* d_in, const int* in_sizes, int n_in,
                              void* d_out, int out_size, void* d_ws, size_t ws_size,
                              hipStream_t stream) {
  const float* x_tasks = (const float*)d_in[0];
  const float* x_data  = (const float*)d_in[1];
  const int*   ei_dt   = (const int*)d_in[2];   // row0 = src(data), row1 = dst(task)
  const int*   ei_td   = (const int*)d_in[3];   // row0 = src(task), row1 = dst(data)
  const float* ea_dt   = (const float*)d_in[4];
  const float* ea_td   = (const float*)d_in[5];
  const float* ln_t_g  = (const float*)d_in[38];
  const float* ln_t_b  = (const float*)d_in[39];
  const float* ln_d_g  = (const float*)d_in[40];
  const float* ln_d_b  = (const float*)d_in[41];

  // --- workspace layout (floats); total ~40.2M floats ~ 161 MB -------------
  float* ws = (float*)d_ws;
  size_t off = 0;
  float* xs_dt = ws + off; off += (size_t)ND * HC;   // xs for *dt convs (src=data/d1)
  float* xs_td = ws + off; off += (size_t)NT * HC;   // xs for *td convs (src=task/t1)
  float* t1    = ws + off; off += (size_t)NT * HC;
  float* d1    = ws + off; off += (size_t)ND * HC;
  float* al_s  = ws + off; off += NT;
  float* al_d  = ws + off; off += NT;
  unsigned* amax = (unsigned*)(ws + off); off += NT;
  float* den   = ws + off; off += NT;
  float* albuf = ws + off; off += EDGES;
  float* wvec  = ws + off; off += 512;               // wda[ci*64], wea[256+ci*4]

  const int B256 = 256;
  const int E = EDGES;

  // --- precompute Wd^T a_d and We^T a_e for the 4 convs --------------------
  const int indK[4] = {TF, DF, HC, HC};              // dst-feature K per conv
  for (int ci = 0; ci < 4; ++ci) {
    int B = 6 + ci * 8;                              // Ws,Wd,We,as,ad,ae,Wr,b
    matTvec_kernel<<<1, 64, 0, stream>>>(
        (const float*)d_in[B + 1], (const float*)d_in[B + 4], wvec + ci * 64, indK[ci]);
    matTvec_kernel<<<1, 64, 0, stream>>>(
        (const float*)d_in[B + 2], (const float*)d_in[B + 5], wvec + 256 + ci * 4, EF);
  }

  auto run_conv = [&](int B, int ci,
                      const float* srcF, int Ns, int Ks,
                      const float* dstF, int Nd, int Kd,
                      const int* esrc, const int* edst, const float* ea,
                      float* xs, float* out) {
    const float* Ws   = (const float*)d_in[B + 0];
    const float* a_s  = (const float*)d_in[B + 3];
    const float* Wr   = (const float*)d_in[B + 6];
    const float* bias = (const float*)d_in[B + 7];
    // xs = src @ Ws^T
    if (Ks == HC)
      gemm64_wmma_kernel<<<Ns / 32, B256, 0, stream>>>(srcF, Ws, (const float*)nullptr, xs, Ns);
    else
      gemm_smallk_kernel<<<(int)(((size_t)Ns * HC + 255) / 256), B256, 0, stream>>>(
          srcF, Ws, (const float*)nullptr, xs, Ns, Ks);
    // out = dst @ Wr^T + b   (residual base; fully initializes out)
    if (Kd == HC)
      gemm64_wmma_kernel<<<Nd / 32, B256, 0, stream>>>(dstF, Wr, bias, out, Nd);
    else
      gemm_smallk_kernel<<<(int)(((size_t)Nd * HC + 255) / 256), B256, 0, stream>>>(
          dstF, Wr, bias, out, Nd, Kd);
    // attention scalar projections
    dot_rows_kernel<<<(Ns + 255) / 256, B256, 0, stream>>>(xs, a_s, al_s, Ns, HC);
    dot_rows_kernel<<<(Nd + 255) / 256, B256, 0, stream>>>(dstF, wvec + ci * 64, al_d, Nd, Kd);
    // segment softmax + aggregation
    init_softmax_kernel<<<(Nd + 255) / 256, B256, 0, stream>>>(amax, den, Nd);
    edge_attn_max_kernel<<<(E + 255) / 256, B256, 0, stream>>>(
        esrc, edst, ea, al_s, al_d, wvec + 256 + ci * 4, albuf, amax, E);
    amax_decode_kernel<<<(Nd + 255) / 256, B256, 0, stream>>>(amax, Nd);
    edge_exp_kernel<<<(E + 255) / 256, B256, 0, stream>>>(
        edst, albuf, (const float*)amax, den, E);
    edge_agg_kernel<<<(int)(((size_t)E * 16 + 255) / 256), B256, 0, stream>>>(
        esrc, edst, albuf, den, xs, out, E);
  };

  // --- layer 1 -------------------------------------------------------------
  // t1 = GAT(data -> tasks), d1 = GAT(tasks -> data)
  run_conv(6, 0, x_data, ND, DF, x_tasks, NT, TF, ei_dt, ei_dt + E, ea_dt, xs_dt, t1);
  run_conv(14, 1, x_tasks, NT, TF, x_data, ND, DF, ei_td, ei_td + E, ea_td, xs_td, d1);
  act_ln_kernel<<<(int)(((size_t)NT * 32 + 255) / 256), B256, 0, stream>>>(t1, ln_t_g, ln_t_b, NT);
  act_ln_kernel<<<(int)(((size_t)ND * 32 + 255) / 256), B256, 0, stream>>>(d1, ln_d_g, ln_d_b, ND);

  // --- layer 2 (K=64 GEMMs run on WMMA; outputs go straight into d_out) ----
  float* t2 = (float*)d_out;                       // [NT, 64]
  float* d2 = (float*)d_out + (size_t)NT * HC;     // [ND, 64]
  run_conv(22, 2, t1, NT, HC, d1, ND, HC, ei_td, ei_td + E, ea_td, xs_td, d2);
  run_conv(30, 3, d1, ND, HC, t1, NT, HC, ei_dt, ei_dt + E, ea_dt, xs_dt, t2);
}